// OntologyNNC_70497593197362
// MI455X (gfx1250) — compile-verified
//
#include <hip/hip_runtime.h>
#include <math.h>

#define B_   32
#define F_   200
#define N_   50000
#define E_   800000
#define C_   3
#define OUT_ 4
#define E2_  (E_ + N_)
#define NB_  (N_ * 32)

typedef __attribute__((ext_vector_type(2))) float v2f;
typedef __attribute__((ext_vector_type(8))) float v8f;

// ---------------------------------------------------------------------------
// Native CDNA5 non-returning FP atomics (GLOBAL_ATOMIC_ADD_F32 /
// GLOBAL_ATOMIC_MAX_NUM_F32, ISA §15.18.3 op 86/82). Fire-and-forget: tracked
// on STOREcnt, drained by the implicit wait at S_ENDPGM. Device scope so the
// reduction is coherent across WGPs. Avoids clang's CAS-loop expansion.
// ---------------------------------------------------------------------------
__device__ __forceinline__ void atom_add_f32(float* p, float v) {
    asm volatile("global_atomic_add_f32 %0, %1, off scope:SCOPE_DEV"
                 :: "v"(p), "v"(v) : "memory");
}
__device__ __forceinline__ void atom_max_f32(float* p, float v) {
    asm volatile("global_atomic_max_num_f32 %0, %1, off scope:SCOPE_DEV"
                 :: "v"(p), "v"(v) : "memory");
}

// ---------------------------------------------------------------------------
// Kernel 1: x = feature_data @ fc1_w.T + fc1_b ; h = x * gat_w
// M=32 (batches), K=200, N=50000 (nodes). One wave computes a 32x16 tile via
// V_WMMA_F32_16X16X4_F32 (two WMMAs per K=4 step: batches 0-15 and 16-31).
// 32-bit A 16x4 layout: VGPR0 = K {0|2}, VGPR1 = K {1|3} split at lane 16.
// Output stored node-major h[n*32 + b] so edge waves get coalesced 128B rows.
// ---------------------------------------------------------------------------
__global__ void k_fc1_wmma(const float* __restrict__ fd,     // [32,200]
                           const float* __restrict__ w,      // [N,200]
                           const float* __restrict__ bias,   // [N]
                           const float* __restrict__ gatw,   // [1]
                           float* __restrict__ h)            // [N,32]
{
    const int wave = threadIdx.x >> 5;
    const int tile = blockIdx.x * 8 + wave;          // 16-node tile id
    if (tile >= (N_ / 16)) return;                   // wave-uniform guard
    const int lane = threadIdx.x & 31;
    const int half = lane >> 4;                      // 0: K+{0,1}, 1: K+{2,3}
    const int l16  = lane & 15;
    const int n0   = tile * 16;
    const float gw = gatw[0];

    v8f c0 = {};   // batches 0..15
    v8f c1 = {};   // batches 16..31
    const float* fd0 = fd + (size_t)l16 * F_;
    const float* fd1 = fd + (size_t)(16 + l16) * F_;
    const float* wr  = w  + (size_t)(n0 + l16) * F_;

    for (int k = 0; k < F_; k += 4) {
        const int ka = k + 2 * half;
        v2f a0, a1, bb;
        a0.x = fd0[ka]; a0.y = fd0[ka + 1];
        a1.x = fd1[ka]; a1.y = fd1[ka + 1];
        bb.x = wr[ka];  bb.y = wr[ka + 1];
        c0 = __builtin_amdgcn_wmma_f32_16x16x4_f32(false, a0, false, bb,
                                                   (short)0, c0, false, false);
        c1 = __builtin_amdgcn_wmma_f32_16x16x4_f32(false, a1, false, bb,
                                                   (short)0, c1, false, false);
    }

    const float bn = bias[n0 + l16];
    float* hrow = h + (size_t)(n0 + l16) * 32;
    // C/D 16x16 layout: VGPR r -> (M = r + 8*half, N = l16)
#pragma unroll
    for (int r = 0; r < 8; ++r) {
        const int m = r + 8 * half;
        hrow[m]      = (c0[r] + bn) * gw;
        hrow[m + 16] = (c1[r] + bn) * gw;
    }
}

// ---------------------------------------------------------------------------
// Init segment buffers
// ---------------------------------------------------------------------------
__global__ void k_init(float* __restrict__ m, float* __restrict__ ssum,
                       float* __restrict__ outg, float* __restrict__ sacc)
{
    const int i = blockIdx.x * blockDim.x + threadIdx.x;
    if (i < NB_) {
        m[i]    = -INFINITY;
        ssum[i] = 0.f;
        outg[i] = 0.f;
        sacc[i] = 0.f;
    }
}

// ---------------------------------------------------------------------------
// Edge helpers: one wave per edge, lane = batch index (wave32 sweet spot).
// src2/dst2 = concat(edge_index rows, self-loops).
// ---------------------------------------------------------------------------
__device__ __forceinline__ void edge_sd(const int* __restrict__ ei, int e,
                                        int& s, int& d)
{
    if (e < E_) { s = ei[e]; d = ei[E_ + e]; }
    else        { s = e - E_; d = s; }
}
__device__ __forceinline__ float lrelu02(float x) { return x > 0.f ? x : 0.2f * x; }

__global__ void k_edge_max(const int* __restrict__ ei, const float* __restrict__ h,
                           const float* __restrict__ as_, const float* __restrict__ ad_,
                           float* __restrict__ mbuf)
{
    const int e = blockIdx.x * 8 + (threadIdx.x >> 5);
    if (e >= E2_) return;
    const int lane = threadIdx.x & 31;
    int s, d; edge_sd(ei, e, s, d);
    const float ev = lrelu02(as_[0] * h[(size_t)s * 32 + lane] +
                             ad_[0] * h[(size_t)d * 32 + lane]);
    atom_max_f32(&mbuf[(size_t)d * 32 + lane], ev);
}

__global__ void k_edge_expsum(const int* __restrict__ ei, const float* __restrict__ h,
                              const float* __restrict__ as_, const float* __restrict__ ad_,
                              const float* __restrict__ mbuf, float* __restrict__ ssum)
{
    const int e = blockIdx.x * 8 + (threadIdx.x >> 5);
    if (e >= E2_) return;
    const int lane = threadIdx.x & 31;
    int s, d; edge_sd(ei, e, s, d);
    const size_t di = (size_t)d * 32 + lane;
    const float ev = lrelu02(as_[0] * h[(size_t)s * 32 + lane] + ad_[0] * h[di]);
    const float ex = __expf(ev - mbuf[di]);
    atom_add_f32(&ssum[di], ex);
}

__global__ void k_edge_gat(const int* __restrict__ ei, const float* __restrict__ h,
                           const float* __restrict__ as_, const float* __restrict__ ad_,
                           const float* __restrict__ mbuf, const float* __restrict__ ssum,
                           float* __restrict__ outg)
{
    const int e = blockIdx.x * 8 + (threadIdx.x >> 5);
    if (e >= E2_) return;
    const int lane = threadIdx.x & 31;
    int s, d; edge_sd(ei, e, s, d);
    const size_t si = (size_t)s * 32 + lane;
    const size_t di = (size_t)d * 32 + lane;
    const float hs = h[si];
    const float ev = lrelu02(as_[0] * hs + ad_[0] * h[di]);
    const float alpha = __expf(ev - mbuf[di]) / ssum[di];
    atom_add_f32(&outg[di], alpha * hs);
}

// x_enc = elu(out_gat + gat_b)
__global__ void k_xenc(const float* __restrict__ outg, const float* __restrict__ gatb,
                       float* __restrict__ xenc)
{
    const int i = blockIdx.x * blockDim.x + threadIdx.x;
    if (i < NB_) {
        const float v = outg[i] + gatb[0];
        xenc[i] = (v > 0.f) ? v : (__expf(v) - 1.f);
    }
}

// s[b,n] = segsum(alpha * x_enc[src])   (deg==1 -> norm==alpha exactly)
__global__ void k_edge_gcn(const int* __restrict__ ei, const float* __restrict__ h,
                           const float* __restrict__ as_, const float* __restrict__ ad_,
                           const float* __restrict__ mbuf, const float* __restrict__ ssum,
                           const float* __restrict__ xenc, float* __restrict__ sacc)
{
    const int e = blockIdx.x * 8 + (threadIdx.x >> 5);
    if (e >= E2_) return;
    const int lane = threadIdx.x & 31;
    int s, d; edge_sd(ei, e, s, d);
    const size_t si = (size_t)s * 32 + lane;
    const size_t di = (size_t)d * 32 + lane;
    const float ev = lrelu02(as_[0] * h[si] + ad_[0] * h[di]);
    const float alpha = __expf(ev - mbuf[di]) / ssum[di];
    atom_add_f32(&sacc[di], alpha * xenc[si]);
}

// ---------------------------------------------------------------------------
// comm path: agg[c] = s*gcn_w[c] + gcn_b[c]; leaky(0.01); LayerNorm(C=3);
// softmax(C=3) -> d_out[128 + b*N*3 + n*3 + c]
// ---------------------------------------------------------------------------
__global__ void k_comm(const float* __restrict__ sacc, const float* __restrict__ gcnw,
                       const float* __restrict__ gcnb, const float* __restrict__ lng,
                       const float* __restrict__ lnb, float* __restrict__ out)
{
    const int n = blockIdx.x * blockDim.x + threadIdx.x;
    const int b = blockIdx.y;
    if (n >= N_) return;
    const float sv = sacc[(size_t)n * 32 + b];
    float v[C_];
#pragma unroll
    for (int c = 0; c < C_; ++c) {
        const float a = sv * gcnw[c] + gcnb[c];
        v[c] = (a > 0.f) ? a : 0.01f * a;
    }
    const float mu  = (v[0] + v[1] + v[2]) * (1.f / 3.f);
    const float d0 = v[0] - mu, d1 = v[1] - mu, d2 = v[2] - mu;
    const float var = (d0 * d0 + d1 * d1 + d2 * d2) * (1.f / 3.f);
    const float rs = rsqrtf(var + 1e-5f);
    float t[C_], mx = -INFINITY;
#pragma unroll
    for (int c = 0; c < C_; ++c) {
        t[c] = (v[c] - mu) * rs * lng[c] + lnb[c];
        mx = fmaxf(mx, t[c]);
    }
    float sum = 0.f;
#pragma unroll
    for (int c = 0; c < C_; ++c) { t[c] = __expf(t[c] - mx); sum += t[c]; }
    const float inv = 1.f / sum;
    float* o = out + 128 + (size_t)b * N_ * C_ + (size_t)n * C_;
#pragma unroll
    for (int c = 0; c < C_; ++c) o[c] = t[c] * inv;
}

// ---------------------------------------------------------------------------
// Classifier: logits[b,:] = x_enc[b,:] @ clf_w.T + clf_b; softmax over OUT=4.
// One block per batch; 256-thread strided reduction, LDS tree.
// ---------------------------------------------------------------------------
__global__ void k_clf(const float* __restrict__ xenc, const float* __restrict__ cw,
                      const float* __restrict__ cb, float* __restrict__ out)
{
    __shared__ float red[OUT_ * 256];
    const int b = blockIdx.x;
    const int t = threadIdx.x;
    float p[OUT_] = {0.f, 0.f, 0.f, 0.f};
    for (int n = t; n < N_; n += 256) {
        const float xe = xenc[(size_t)n * 32 + b];
#pragma unroll
        for (int o = 0; o < OUT_; ++o) p[o] += xe * cw[(size_t)o * N_ + n];
    }
#pragma unroll
    for (int o = 0; o < OUT_; ++o) red[o * 256 + t] = p[o];
    __syncthreads();
    for (int st = 128; st > 0; st >>= 1) {
        if (t < st) {
#pragma unroll
            for (int o = 0; o < OUT_; ++o) red[o * 256 + t] += red[o * 256 + t + st];
        }
        __syncthreads();
    }
    if (t == 0) {
        float l[OUT_], mx = -INFINITY;
#pragma unroll
        for (int o = 0; o < OUT_; ++o) { l[o] = red[o * 256] + cb[o]; mx = fmaxf(mx, l[o]); }
        float sum = 0.f;
#pragma unroll
        for (int o = 0; o < OUT_; ++o) { l[o] = __expf(l[o] - mx); sum += l[o]; }
        const float inv = 1.f / sum;
#pragma unroll
        for (int o = 0; o < OUT_; ++o) out[b * OUT_ + o] = l[o] * inv;
    }
}

// ---------------------------------------------------------------------------
extern "C" void kernel_launch(void* const* d_in, const int* in_sizes, int n_in,
                              void* d_out, int out_size, void* d_ws, size_t ws_size,
                              hipStream_t stream)
{
    const float* fd   = (const float*)d_in[0];
    const float* fc1w = (const float*)d_in[1];
    const float* fc1b = (const float*)d_in[2];
    const float* gatw = (const float*)d_in[3];
    const float* atts = (const float*)d_in[4];
    const float* attd = (const float*)d_in[5];
    const float* gatb = (const float*)d_in[6];
    const float* gcnw = (const float*)d_in[7];
    const float* gcnb = (const float*)d_in[8];
    const float* lng  = (const float*)d_in[9];
    const float* lnb  = (const float*)d_in[10];
    const float* clfw = (const float*)d_in[11];
    const float* clfb = (const float*)d_in[12];
    const int*   ei   = (const int*)d_in[13];
    float* out = (float*)d_out;

    float* ws   = (float*)d_ws;       // 6 x [N,32] f32 = 38.4 MB (L2-resident)
    float* h    = ws;
    float* mbuf = ws + 1 * (size_t)NB_;
    float* ssum = ws + 2 * (size_t)NB_;
    float* outg = ws + 3 * (size_t)NB_;
    float* xenc = ws + 4 * (size_t)NB_;
    float* sacc = ws + 5 * (size_t)NB_;

    const int gemm_blocks = (N_ / 16 + 7) / 8;          // 8 waves/block
    k_fc1_wmma<<<gemm_blocks, 256, 0, stream>>>(fd, fc1w, fc1b, gatw, h);

    k_init<<<(NB_ + 255) / 256, 256, 0, stream>>>(mbuf, ssum, outg, sacc);

    const int eb = (E2_ + 7) / 8;                       // 8 edges (waves)/block
    k_edge_max   <<<eb, 256, 0, stream>>>(ei, h, atts, attd, mbuf);
    k_edge_expsum<<<eb, 256, 0, stream>>>(ei, h, atts, attd, mbuf, ssum);
    k_edge_gat   <<<eb, 256, 0, stream>>>(ei, h, atts, attd, mbuf, ssum, outg);
    k_xenc<<<(NB_ + 255) / 256, 256, 0, stream>>>(outg, gatb, xenc);
    k_edge_gcn   <<<eb, 256, 0, stream>>>(ei, h, atts, attd, mbuf, ssum, xenc, sacc);

    dim3 gcomm((N_ + 255) / 256, B_);
    k_comm<<<gcomm, 256, 0, stream>>>(sacc, gcnw, gcnb, lng, lnb, out);

    k_clf<<<B_, 256, 0, stream>>>(xenc, clfw, clfb, out);
}